// RandomShiftsAug_25658134626510
// MI455X (gfx1250) — compile-verified
//
#include <hip/hip_runtime.h>

#define PAD   4
#define H     84
#define W     84
#define PLANE (H * W)      // 7056 floats = 28224 bytes per (n,c) plane
#define VEC4S (PLANE / 4)  // 1764 float4 chunks (W % 4 == 0 -> each chunk within one row)
#define TPB   256

typedef __attribute__((ext_vector_type(4))) int   v4i;
typedef __attribute__((ext_vector_type(4))) float v4f;
typedef __attribute__((address_space(1))) v4i* global_v4i_ptr;
typedef __attribute__((address_space(3))) v4i* lds_v4i_ptr;

__global__ __launch_bounds__(TPB)
void RandomShiftsAug_kernel(const float* __restrict__ x,
                            const int*   __restrict__ shift,
                            float*       __restrict__ out)
{
    __shared__ float tile[PLANE];   // full source plane staged in LDS

    const int ch  = blockIdx.x;                    // channel
    const int n   = blockIdx.y;                    // image id (shift is per image)
    const int nc  = n * gridDim.x + ch;            // flat plane id
    const int tid = threadIdx.x;

    const float* __restrict__ src = x   + (size_t)nc * PLANE;
    float*       __restrict__ dst = out + (size_t)nc * PLANE;

    // ---- Stage the whole 84x84 plane into LDS via async global->LDS B128 ----
    for (int i = tid; i < VEC4S; i += TPB) {
        float* g = const_cast<float*>(src + i * 4);
        float* l = &tile[i * 4];
#if __has_builtin(__builtin_amdgcn_global_load_async_to_lds_b128)
        __builtin_amdgcn_global_load_async_to_lds_b128(
            (global_v4i_ptr)g,
            (lds_v4i_ptr)l,
            /*offset=*/0, /*cpol=*/0);
#else
        unsigned lds_off = (unsigned)(size_t)l; // low 32 bits = LDS offset
        asm volatile("global_load_async_to_lds_b128 %0, %1, off"
                     :: "v"(lds_off), "v"(g) : "memory");
#endif
    }

    // Wait for this wave's async copies, then make LDS visible to all waves.
#if __has_builtin(__builtin_amdgcn_s_wait_asynccnt)
    __builtin_amdgcn_s_wait_asynccnt(0);
#else
    asm volatile("s_wait_asynccnt 0" ::: "memory");
#endif
    __syncthreads();

    // Per-image shift (uniform across block -> scalar loads)
    const int ox = shift[2 * n + 0] - PAD;  // column shift
    const int oy = shift[2 * n + 1] - PAD;  // row shift

    // ---- Gather from LDS with clamped indices; stream out as aligned float4 ----
    for (int i = tid; i < VEC4S; i += TPB) {
        const int e0  = i * 4;
        const int r   = e0 / W;           // constant divisor -> mul/shift
        const int col = e0 - r * W;

        const int sr = min(max(r + oy, 0), H - 1);
        const float* __restrict__ row = &tile[sr * W];

        const int c0 = min(max(col + 0 + ox, 0), W - 1);
        const int c1 = min(max(col + 1 + ox, 0), W - 1);
        const int c2 = min(max(col + 2 + ox, 0), W - 1);
        const int c3 = min(max(col + 3 + ox, 0), W - 1);

        v4f v;
        v.x = row[c0];
        v.y = row[c1];
        v.z = row[c2];
        v.w = row[c3];

        // Non-temporal 16B store: write-once 260MB stream, keep it out of L2's way.
        __builtin_nontemporal_store(v, reinterpret_cast<v4f*>(dst + e0));
    }
}

extern "C" void kernel_launch(void* const* d_in, const int* in_sizes, int n_in,
                              void* d_out, int out_size, void* d_ws, size_t ws_size,
                              hipStream_t stream)
{
    const float* x     = (const float*)d_in[0];   // (B, N, c, h, w) f32
    const int*   shift = (const int*)d_in[1];     // (B*N, 2) i32

    float* out = (float*)d_out;

    const int n_img     = in_sizes[1] / 2;                  // B*N = 1024
    const int c_per_img = in_sizes[0] / (n_img * PLANE);    // 9

    dim3 grid(c_per_img, n_img);                            // 9 x 1024 planes
    RandomShiftsAug_kernel<<<grid, TPB, 0, stream>>>(x, shift, out);
}